// SSTv2_24687472017457
// MI455X (gfx1250) — compile-verified
//
#include <hip/hip_runtime.h>

// ---------------------------------------------------------------------------
// SSTv2 linear0 + recover_bev fused for MI455X (gfx1250, wave32, WMMA)
//   feat = voxel_feats @ W0^T + b0        (fp32 GEMM via v_wmma_f32_16x16x4_f32)
//   canvas[b, c, y, x] = feat[n, c]       (scatter, canvas zero elsewhere)
// ---------------------------------------------------------------------------

typedef __attribute__((ext_vector_type(2))) float v2f;
typedef __attribute__((ext_vector_type(8))) float v8f;

#define NVOX    200000
#define CIN     128
#define DM      128
#define NYD     400
#define NXD     400
#define BATCH   4
#define HW      (NYD * NXD)            // 160000
#define NBLK    (NVOX / 16)            // 12500 row-blocks of 16 voxels
#define A_STRIDE 132                   // 128 + 4 pad floats -> conflict-free LDS

// ---------------------------------------------------------------------------
// Zero-fill canvas: 81.92M floats with coalesced b128 stores.
// ---------------------------------------------------------------------------
__global__ __launch_bounds__(256) void sst_zero(float4* __restrict__ out, int n4) {
  int i = blockIdx.x * blockDim.x + threadIdx.x;
  const int stride = gridDim.x * blockDim.x;
  float4 z; z.x = z.y = z.z = z.w = 0.0f;
  for (; i < n4; i += stride) out[i] = z;
}

// ---------------------------------------------------------------------------
// Fused WMMA GEMM + batch-aware BEV scatter.
// 256 threads = 8 wave32; wave w owns output-channel tile [16w, 16w+16).
// ---------------------------------------------------------------------------
__global__ __launch_bounds__(256) void sst_gemm_scatter(
    const float* __restrict__ vf,     // [NVOX, CIN]
    const float* __restrict__ W0,     // [DM, CIN]
    const float* __restrict__ b0,     // [DM]
    const int*   __restrict__ coors,  // [NVOX, 4] (b, z, y, x) int32
    float*       __restrict__ out)    // [BATCH, DM, NYD, NXD]
{
  __shared__ float Alds[16 * A_STRIDE];  // 16-voxel x 128-feat panel (padded)
  __shared__ int   Base[16];             // per-voxel canvas base offset

  const int tid  = threadIdx.x;
  const int wave = tid >> 5;             // 0..7: channel tile
  const int lane = tid & 31;
  const int half = lane >> 4;            // lane half selects K pair
  const int lm   = lane & 15;
  const int d    = wave * 16 + lm;       // output channel (N) owned by lane

  // Preload this wave's full-K B operand (W0^T tile) into registers once.
  // B layout for 16x16x4 f32: lane L holds N = L%16; lane half selects K=0,1 / 2,3.
  v2f Breg[32];
  #pragma unroll
  for (int kk = 0; kk < 32; ++kk) {
    Breg[kk] = *(const v2f*)(W0 + d * CIN + kk * 4 + half * 2);
  }
  const float bias = b0[d];

  for (int blk = blockIdx.x; blk < NBLK; blk += gridDim.x) {
    const int row0 = blk * 16;

    __syncthreads();  // protect LDS reuse across grid-stride iterations

    // Stage A panel: 2048 floats, 256 threads x 2 float4 loads/stores.
    {
      const int r  = tid >> 4;           // voxel row 0..15
      const int c0 = (tid & 15) * 8;     // feat col 0..120
      const float4* src = (const float4*)(vf + (size_t)(row0 + r) * CIN + c0);
      float4 u0 = src[0];
      float4 u1 = src[1];
      *(float4*)&Alds[r * A_STRIDE + c0]     = u0;
      *(float4*)&Alds[r * A_STRIDE + c0 + 4] = u1;
    }
    // Per-voxel scatter base: b*(C*HW) + y*NX + x  (coors row = {b, z, y, x})
    if (tid < 16) {
      const int n = row0 + tid;
      const int b = coors[4 * n + 0];
      const int y = coors[4 * n + 2];
      const int x = coors[4 * n + 3];
      Base[tid] = b * (DM * HW) + y * NXD + x;
    }
    __syncthreads();

    // 16x16 output tile: D = A(16x128) x B(128x16), 32 steps of K=4.
    v8f acc = {};
    #pragma unroll
    for (int kk = 0; kk < 32; ++kk) {
      // A layout: lane L holds M = L%16; lane half selects K pair within step.
      v2f a = *(const v2f*)&Alds[lm * A_STRIDE + kk * 4 + half * 2];
      acc = __builtin_amdgcn_wmma_f32_16x16x4_f32(
          /*neg_a=*/false, a, /*neg_b=*/false, Breg[kk],
          /*c_mod=*/(short)0, acc, /*reuse_a=*/false, /*reuse_b=*/false);
    }

    // Scatter: C/D layout — VGPR v holds M = v (lanes 0-15) / v+8 (lanes 16-31),
    // N = lane%16. Each lane writes 8 channel-d values for 8 distinct voxels.
    const int dbase = d * HW;
    #pragma unroll
    for (int v = 0; v < 8; ++v) {
      const int m = v + half * 8;              // voxel row within block
      out[Base[m] + dbase] = acc[v] + bias;
    }
  }
}

// ---------------------------------------------------------------------------
// Launch wrapper
// ---------------------------------------------------------------------------
extern "C" void kernel_launch(void* const* d_in, const int* in_sizes, int n_in,
                              void* d_out, int out_size, void* d_ws, size_t ws_size,
                              hipStream_t stream) {
  const float* vf    = (const float*)d_in[0];  // voxel_feats [N, 128]
  const float* W0    = (const float*)d_in[1];  // [128, 128]
  const float* b0    = (const float*)d_in[2];  // [128]
  const int*   coors = (const int*)d_in[3];    // [N, 4] int32 (JAX x64 disabled)
  float* out = (float*)d_out;                  // [4, 128, 400, 400]

  const int n4 = (BATCH * DM * HW) / 4;        // 20,480,000 float4
  sst_zero<<<4096, 256, 0, stream>>>((float4*)out, n4);
  sst_gemm_scatter<<<2048, 256, 0, stream>>>(vf, W0, b0, coors, out);
}